// linear_coding_50345606644290
// MI455X (gfx1250) — compile-verified
//
#include <hip/hip_runtime.h>

typedef __attribute__((ext_vector_type(2))) float v2f;
typedef __attribute__((ext_vector_type(8))) float v8f;

#define BATCH      8192
#define SPARSITY   64
#define BASIS_NUM  4096
#define EMB_DIM    512

#define ROWS_PER_BLK 16
#define SEL_STRIDE   516    // padded LDS row stride (floats) for gathered A tile
#define SC_STRIDE    4100   // padded LDS row stride (floats) for score strip

// LDS budget: 16*516*4 + 16*4100*4 + 16*64*4 = 299,520 B  (< 320 KB WGP LDS)
#define LDS_BYTES ((ROWS_PER_BLK*SEL_STRIDE + ROWS_PER_BLK*SC_STRIDE) * 4 + ROWS_PER_BLK*SPARSITY*4)

// ---------------- Kernel 1: bsq[n] = ||basis_n||^2 -------------------------
__global__ void bsq_kernel(const float* __restrict__ basis, float* __restrict__ bsq) {
  const int wave = threadIdx.x >> 5;           // 8 waves / block, one row each
  const int lane = threadIdx.x & 31;
  const int row  = blockIdx.x * 8 + wave;
  if (row >= BASIS_NUM) return;
  const float4* r = (const float4*)(basis + (long)row * EMB_DIM);
  float s = 0.f;
#pragma unroll
  for (int i = 0; i < 4; ++i) {                // 128 float4 per row / 32 lanes
    float4 v = r[lane + 32 * i];
    s += v.x * v.x + v.y * v.y + v.z * v.z + v.w * v.w;
  }
#pragma unroll
  for (int off = 16; off > 0; off >>= 1) s += __shfl_xor(s, off, 32);
  if (lane == 0) bsq[row] = s;
}

// ---------------- Kernel 2: fused score-GEMM + top-64 + weighted gather ----
extern __shared__ float smem[];

__global__ void __launch_bounds__(256, 1)
lincode_kernel(const float* __restrict__ x, const float* __restrict__ basis,
               const int* __restrict__ index, const float* __restrict__ bsq,
               float* __restrict__ out) {
  float* selA = smem;                                      // [16][516]
  float* sc   = smem + ROWS_PER_BLK * SEL_STRIDE;          // [16][4100]
  int*   topI = (int*)(sc + ROWS_PER_BLK * SC_STRIDE);     // [16][64]

  const int b0   = blockIdx.x * ROWS_PER_BLK;
  const int tid  = threadIdx.x;
  const int wave = tid >> 5;                               // 8 waves
  const int lane = tid & 31;

  // ---- Phase 1: gather A tile (sel rows) into LDS --------------------------
  {
    const int r   = tid >> 4;                              // 0..15 row
    const int c16 = tid & 15;
    const int srow = index[b0 + r];
    const float4* src = (const float4*)(basis + (long)srow * EMB_DIM);
#pragma unroll
    for (int ch = 0; ch < 8; ++ch) {
      const int col4 = c16 + ch * 16;                      // float4 index 0..127
      float4 v = src[col4];
      *(float4*)(selA + r * SEL_STRIDE + col4 * 4) = v;
    }
  }
  __syncthreads();

  // ---- Phase 2: score GEMM via V_WMMA_F32_16X16X4_F32 ----------------------
  // A frag (16x4 f32): lanes 0-15 -> M=lane, K={k0,k0+1}; lanes 16-31 -> K={k0+2,k0+3}
  // C/D (16x16 f32): element (M,N): vgpr = M%8, lane = 16*(M/8) + N
  {
    const int m  = lane & 15;
    const int hi = lane >> 4;
    const float* aBase = selA + m * SEL_STRIDE + 2 * hi;
#pragma unroll 1
    for (int t = 0; t < 32; ++t) {                         // 32 N-tiles per wave
      const int n0 = wave * 512 + t * 16;
      const float* bRow = basis + (long)(n0 + m) * EMB_DIM + 2 * hi;
      v8f acc = {};
      for (int k0 = 0; k0 < EMB_DIM; k0 += 4) {
        v2f a, b;
        a.x = aBase[k0];  a.y = aBase[k0 + 1];             // ds_load from LDS
        b.x = bRow[k0];   b.y = bRow[k0 + 1];              // global (L2-resident)
        acc = __builtin_amdgcn_wmma_f32_16x16x4_f32(
            false, a, false, b, (short)0, acc, false, false);
      }
      const float q = bsq[n0 + m];                          // ||basis_n||^2
#pragma unroll
      for (int r = 0; r < 8; ++r) {
        const int mm = r + 8 * hi;
        sc[mm * SC_STRIDE + n0 + m] = q - 2.0f * acc[r];    // score to minimize
      }
    }
  }
  __syncthreads();

  // ---- Phase 3: per-row top-64 smallest (ascending, ties -> lower index) ---
  const float INF = __builtin_inff();
#pragma unroll 1
  for (int rr = 0; rr < 2; ++rr) {                          // 2 rows per wave
    const int row = wave * 2 + rr;
    float* srow = sc + row * SC_STRIDE;
#pragma unroll 1
    for (int j = 0; j < SPARSITY; ++j) {
      float bv = INF;
      int   bi = BASIS_NUM;
      for (int i = lane; i < BASIS_NUM; i += 32) {          // strict '<' keeps lowest idx
        const float v = srow[i];
        if (v < bv) { bv = v; bi = i; }
      }
#pragma unroll
      for (int off = 16; off > 0; off >>= 1) {              // wave32 butterfly
        const float ov = __shfl_xor(bv, off, 32);
        const int   oi = __shfl_xor(bi, off, 32);
        if (ov < bv || (ov == bv && oi < bi)) { bv = ov; bi = oi; }
      }
      if (lane == 0) { topI[row * SPARSITY + j] = bi; srow[bi] = INF; }
    }
  }
  // topI produced and consumed within the same wave; LDS ops are in-order.

  // ---- Phase 4: out[b,:] = sum_j x[b,j] * basis[topI[j], :] ----------------
#pragma unroll 1
  for (int rr = 0; rr < 2; ++rr) {
    const int row = wave * 2 + rr;
    const long g = (long)(b0 + row);
    float4 a0 = {0, 0, 0, 0}, a1 = {0, 0, 0, 0}, a2 = {0, 0, 0, 0}, a3 = {0, 0, 0, 0};
    const float4* bas4 = (const float4*)basis;
#pragma unroll 1
    for (int j = 0; j < SPARSITY; ++j) {
      const int   idx = topI[row * SPARSITY + j];           // LDS broadcast
      const float w   = x[g * SPARSITY + j];                // uniform global load
      const float4* br = bas4 + (long)idx * (EMB_DIM / 4);
      float4 v;
      v = br[lane];      a0.x += w * v.x; a0.y += w * v.y; a0.z += w * v.z; a0.w += w * v.w;
      v = br[lane + 32]; a1.x += w * v.x; a1.y += w * v.y; a1.z += w * v.z; a1.w += w * v.w;
      v = br[lane + 64]; a2.x += w * v.x; a2.y += w * v.y; a2.z += w * v.z; a2.w += w * v.w;
      v = br[lane + 96]; a3.x += w * v.x; a3.y += w * v.y; a3.z += w * v.z; a3.w += w * v.w;
    }
    float4* o4 = (float4*)out + g * (EMB_DIM / 4);
    o4[lane]      = a0;
    o4[lane + 32] = a1;
    o4[lane + 64] = a2;
    o4[lane + 96] = a3;
  }
}

// ---------------------------- launcher -------------------------------------
extern "C" void kernel_launch(void* const* d_in, const int* in_sizes, int n_in,
                              void* d_out, int out_size, void* d_ws, size_t ws_size,
                              hipStream_t stream) {
  const float* xp     = (const float*)d_in[0];   // [8192, 64]
  const float* basis  = (const float*)d_in[1];   // [4096, 512]
  const int*   index  = (const int*)d_in[2];     // [8192]
  float*       out    = (float*)d_out;           // [8192, 512]
  float*       bsq    = (float*)d_ws;            // 4096 floats scratch

  bsq_kernel<<<BASIS_NUM / 8, 256, 0, stream>>>(basis, bsq);
  lincode_kernel<<<BATCH / ROWS_PER_BLK, 256, LDS_BYTES, stream>>>(
      xp, basis, index, bsq, out);
}